// GraphNetwork_50105088475519
// MI455X (gfx1250) — compile-verified
//
#include <hip/hip_runtime.h>
#include <stdint.h>

// ---------------- problem constants (match reference) ----------------
#define NPTS_F   64        // feature dim F
#define BGR      4         // graphs in batch
#define DDIM     8         // voxels per dim (static bound)
#define MMAX     1024      // static bound on clusters
#define VVOX     (BGR*DDIM*DDIM*DDIM)   // 2048
#define NTILES   (MMAX/16)              // 64 centroid tiles
#define POOLRAD  0.2f
#define BIGF     3.0e38f

// ---------------- workspace layout (uint32 element offsets) ----------
enum : int {
  OFF_CNT    = 0,                      // 2048  voxel counts (u32)
  OFF_SUM    = 2048,                   // 6144  voxel pos sums (f32)
  OFF_MIN    = 8192,                   // 3     encoded min(pos) (u32)
  OFF_MS     = 8196,                   // 4     per-batch first cluster m
  OFF_ME     = 8200,                   // 4     per-batch last cluster m+1
  OFF_IDX    = 8204,                   // 1024  compacted voxel ids
  OFF_NEWB   = 9228,                   // 1024  cluster batch (i32)
  OFF_BP     = 10252,                  // 64*96 packed B tiles + col meta (f32)
  OFF_ASSIGN = 16396,                  // N     point->cluster (i32)
  OFF_ENC    = 116396,                 // 1024*64 encoded feature max (u32)
};                                      // total ~728 KB

// monotonic encodings: enc() is strictly increasing in float order
__device__ __forceinline__ uint32_t enc_f(float x) {
  uint32_t u = __float_as_uint(x);
  return (u >> 31) ? ~u : (u | 0x80000000u);
}
__device__ __forceinline__ float dec_f(uint32_t e) {
  uint32_t u = (e >> 31) ? (e ^ 0x80000000u) : ~e;
  return __uint_as_float(u);
}

typedef __attribute__((ext_vector_type(2))) float v2f;
typedef __attribute__((ext_vector_type(8))) float v8f;

// ---------------- K0: init all accumulators ----------------
__global__ void k_init(uint32_t* ws) {
  int i = blockIdx.x * blockDim.x + threadIdx.x;
  if (i < MMAX * NPTS_F) ws[OFF_ENC + i] = 0x007FFFFFu;       // enc(-inf)
  if (i < VVOX)          ws[OFF_CNT + i] = 0u;
  if (i < VVOX * 3)      ((float*)ws)[OFF_SUM + i] = 0.0f;
  if (i < 3)             ws[OFF_MIN + i] = 0xFFFFFFFFu;       // +inf encoded
  if (i < MMAX)          ws[OFF_IDX + i] = 0u;                // pad -> voxel 0
  if (i < BGR) { ws[OFF_MS + i] = 0x7FFFFFFFu; ws[OFF_ME + i] = 0u; }
}

// ---------------- K1: min(pos) per component ----------------
__global__ void k_min(const float* __restrict__ pos, uint32_t* ws, int n) {
  __shared__ uint32_t sm[3];
  if (threadIdx.x < 3) sm[threadIdx.x] = 0xFFFFFFFFu;
  __syncthreads();
  int i = blockIdx.x * blockDim.x + threadIdx.x;
  if (i < n) {
    atomicMin(&sm[0], enc_f(pos[i*3+0]));
    atomicMin(&sm[1], enc_f(pos[i*3+1]));
    atomicMin(&sm[2], enc_f(pos[i*3+2]));
  }
  __syncthreads();
  if (threadIdx.x < 3) atomicMin(&ws[OFF_MIN + threadIdx.x], sm[threadIdx.x]);
}

// ---------------- K2: voxel histogram + position sums ----------------
__global__ void k_vox(const float* __restrict__ pos, const int* __restrict__ batch,
                      uint32_t* ws, int n) {
  int i = blockIdx.x * blockDim.x + threadIdx.x;
  if (i >= n) return;
  int c[3];
  #pragma unroll
  for (int d = 0; d < 3; ++d) {
    float start = dec_f(ws[OFF_MIN + d]) - 0.5f * POOLRAD;
    int cd = (int)floorf((pos[i*3+d] - start) / POOLRAD);
    c[d] = min(max(cd, 0), DDIM - 1);
  }
  int vox = ((batch[i] * DDIM + c[0]) * DDIM + c[1]) * DDIM + c[2];
  atomicAdd(&ws[OFF_CNT + vox], 1u);
  float* sp = (float*)ws + OFF_SUM + vox * 3;
  atomicAdd(&sp[0], pos[i*3+0]);
  atomicAdd(&sp[1], pos[i*3+1]);
  atomicAdd(&sp[2], pos[i*3+2]);
}

// ---------------- K3: compact voxels, centroids, WMMA B-packing -------
// single block, 1024 threads
__global__ void __launch_bounds__(1024) k_compact(uint32_t* ws, float* out) {
  __shared__ int   s[1024];
  __shared__ float cc[MMAX][4];     // cx, cy, cz, |c|^2
  __shared__ float cmeta[MMAX];     // valid ? (float)batch : -1
  int t = threadIdx.x;

  int f0 = (ws[OFF_CNT + 2*t]     > 0u) ? 1 : 0;
  int f1 = (ws[OFF_CNT + 2*t + 1] > 0u) ? 1 : 0;
  s[t] = f0 + f1;
  __syncthreads();
  // Hillis-Steele inclusive scan over 1024 pair-sums
  for (int off = 1; off < 1024; off <<= 1) {
    int v = s[t];
    int a = (t >= off) ? s[t - off] : 0;
    __syncthreads();
    s[t] = v + a;
    __syncthreads();
  }
  int incl  = s[t];
  int excl  = incl - (f0 + f1);
  int total = s[1023];
  int r0 = excl, r1 = excl + f0;
  if (f0 && r0 < MMAX) ws[OFF_IDX + r0] = (uint32_t)(2*t);
  if (f1 && r1 < MMAX) ws[OFF_IDX + r1] = (uint32_t)(2*t + 1);
  __syncthreads();

  // phase 2: per-cluster outputs (m == t)
  int m   = t;
  int vox = (int)ws[OFF_IDX + m];
  float cnt = (float)ws[OFF_CNT + vox];
  float inv = 1.0f / fmaxf(cnt, 1.0f);
  const float* sp = (const float*)ws + OFF_SUM + vox * 3;
  float cx = sp[0] * inv, cy = sp[1] * inv, cz = sp[2] * inv;
  int nb = vox / (DDIM * DDIM * DDIM);
  int valid = (m < total) ? 1 : 0;

  out[MMAX*NPTS_F + m*3 + 0] = cx;                   // centroids
  out[MMAX*NPTS_F + m*3 + 1] = cy;
  out[MMAX*NPTS_F + m*3 + 2] = cz;
  out[MMAX*NPTS_F + MMAX*3 + m]        = (float)nb;  // new_batch
  out[MMAX*NPTS_F + MMAX*3 + MMAX + m] = valid ? 1.0f : 0.0f;  // valid
  ws[OFF_NEWB + m] = (uint32_t)nb;
  if (valid) {
    atomicMin((int*)&ws[OFF_MS + nb], m);
    atomicMax((int*)&ws[OFF_ME + nb], m + 1);
  }
  cc[m][0] = cx; cc[m][1] = cy; cc[m][2] = cz;
  cc[m][3] = cx*cx + cy*cy + cz*cz;
  cmeta[m] = valid ? (float)nb : -1.0f;
  __syncthreads();

  // pack B tiles in the exact WMMA lane layout, interleaved so each lane
  // fetches its (vgpr0,vgpr1) pair with one b64 load:
  //   tile stride 96: [l*2 + k] -> B(K = k + 2*(l/16), col j = tile*16 + l%16)
  //   where B = (-2cx, -2cy, -2cz, |c|^2);  [64 + l] -> column meta.
  for (int q = t; q < NTILES * 96; q += 1024) {
    int tile = q / 96, r = q % 96;
    float val;
    if (r < 64) {
      int l = r >> 1, k = r & 1;
      int K = k + 2 * (l >> 4);
      int j = tile * 16 + (l & 15);
      val = (K == 3) ? cc[j][3] : (-2.0f * cc[j][K]);
    } else {
      int l = r - 64;
      val = cmeta[tile * 16 + (l & 15)];
    }
    ((float*)ws)[OFF_BP + q] = val;
  }
}

// ---------------- K4: WMMA distance + masked argmin ----------------
// one wave32 per 16 points; each v_wmma_f32_16x16x4_f32 scores 16x16 pairs.
// A row = (px,py,pz,1); B col = (-2cx,-2cy,-2cz,|c|^2) => D = d^2 - |p|^2.
__global__ void __launch_bounds__(256) k_assign(const float* __restrict__ pos,
                                                const int* __restrict__ batch,
                                                const uint32_t* __restrict__ ws,
                                                int* __restrict__ assign, int n) {
  int wave = (blockIdx.x * blockDim.x + threadIdx.x) >> 5;
  int lane = threadIdx.x & 31;
  int base = wave * 16;
  if (base >= n) return;                 // wave-uniform: EXEC stays all-ones

  const float* BP = (const float*)ws + OFF_BP;
  int half = lane >> 4;                  // 0: K0/K1, 1: K2/K3
  int col  = lane & 15;

  // A operand (16x4 f32 layout): lane holds point M=col
  const float* P = pos + (size_t)(base + col) * 3;
  v2f A;
  A.x = (half == 0) ? P[0] : P[2];
  A.y = (half == 0) ? P[1] : 1.0f;

  // batches (as float) of the 8 rows this lane accumulates
  float pbf[8];
  #pragma unroll
  for (int r = 0; r < 8; ++r) pbf[r] = (float)batch[base + r + 8 * half];

  // prune centroid tiles to this wave's batch span (batch is sorted);
  // force the bounds into SGPRs so the tile loop is scalar and EXEC is
  // provably all-ones at every WMMA (ISA 7.12 requirement).
  int bmin = __builtin_amdgcn_readfirstlane(batch[base]);
  int bmax = __builtin_amdgcn_readfirstlane(batch[base + 15]);
  int lo = 0x7FFFFFFF, hi = 0;
  for (int b = bmin; b <= bmax; ++b) {
    lo = min(lo, (int)ws[OFF_MS + b]);
    hi = max(hi, (int)ws[OFF_ME + b]);
  }
  int tLo = __builtin_amdgcn_readfirstlane((lo == 0x7FFFFFFF) ? 0 : (lo >> 4));
  int tHi = __builtin_amdgcn_readfirstlane((hi + 15) >> 4);

  float best[8]; int bidx[8];
  #pragma unroll
  for (int r = 0; r < 8; ++r) { best[r] = BIGF; bidx[r] = 0; }

  for (int tile = tLo; tile < tHi; ++tile) {
    const float* bp = BP + tile * 96;
    v2f Bv = *reinterpret_cast<const v2f*>(bp + lane * 2);  // one b64 load
    float meta = bp[64 + lane];          // valid ? batch(col j) : -1
    v8f C = {0.f,0.f,0.f,0.f,0.f,0.f,0.f,0.f};
    C = __builtin_amdgcn_wmma_f32_16x16x4_f32(
            false, A, false, Bv, (short)0, C, false, false);
    int j = tile * 16 + col;
    #pragma unroll
    for (int r = 0; r < 8; ++r) {
      float dv = (meta == pbf[r]) ? C[r] : BIGF;
      if (dv < best[r]) { best[r] = dv; bidx[r] = j; }
    }
  }

  // argmin across the 16 lanes of each half (columns), ties -> smaller j
  #pragma unroll
  for (int r = 0; r < 8; ++r) {
    float v = best[r]; int id = bidx[r];
    #pragma unroll
    for (int msk = 1; msk < 16; msk <<= 1) {
      float ov = __shfl_xor(v, msk, 32);
      int   oi = __shfl_xor(id, msk, 32);
      if (ov < v || (ov == v && oi < id)) { v = ov; id = oi; }
    }
    bidx[r] = id;
  }
  if (col == 0) {
    #pragma unroll
    for (int r = 0; r < 8; ++r) assign[base + r + 8 * half] = bidx[r];
  }
}

// ---------------- K5: feature scatter-max ----------------
__global__ void k_fmax(const float* __restrict__ x, const int* __restrict__ assign,
                       uint32_t* ws, int n) {
  int id = blockIdx.x * blockDim.x + threadIdx.x;
  if (id >= n * (NPTS_F / 4)) return;
  int pt = id >> 4;
  int fg = (id & 15) * 4;
  const float4 v = *reinterpret_cast<const float4*>(x + (size_t)pt * NPTS_F + fg);
  int a = assign[pt];
  uint32_t* dst = ws + OFF_ENC + (size_t)a * NPTS_F + fg;
  atomicMax(&dst[0], enc_f(v.x));
  atomicMax(&dst[1], enc_f(v.y));
  atomicMax(&dst[2], enc_f(v.z));
  atomicMax(&dst[3], enc_f(v.w));
}

// ---------------- K6: decode x_pool (empty -> 0) ----------------
__global__ void k_fin(const uint32_t* __restrict__ ws, float* __restrict__ out) {
  int i = blockIdx.x * blockDim.x + threadIdx.x;
  if (i >= MMAX * NPTS_F) return;
  float v = dec_f(ws[OFF_ENC + i]);
  out[i] = isfinite(v) ? v : 0.0f;
}

// ---------------- launcher ----------------
extern "C" void kernel_launch(void* const* d_in, const int* in_sizes, int n_in,
                              void* d_out, int out_size, void* d_ws, size_t ws_size,
                              hipStream_t stream) {
  const float* pos   = (const float*)d_in[0];
  const float* x     = (const float*)d_in[1];
  const int*   batch = (const int*)d_in[2];
  float*       out   = (float*)d_out;
  uint32_t*    ws    = (uint32_t*)d_ws;
  int n = in_sizes[2];                           // N points
  int* assign = (int*)(ws + OFF_ASSIGN);

  const int TB = 256;
  k_init<<<(MMAX * NPTS_F + TB - 1) / TB, TB, 0, stream>>>(ws);
  k_min<<<(n + TB - 1) / TB, TB, 0, stream>>>(pos, ws, n);
  k_vox<<<(n + TB - 1) / TB, TB, 0, stream>>>(pos, batch, ws, n);
  k_compact<<<1, 1024, 0, stream>>>(ws, out);
  int nwaves = (n + 15) / 16;
  k_assign<<<(nwaves * 32 + TB - 1) / TB, TB, 0, stream>>>(pos, batch, ws, assign, n);
  k_fmax<<<(n * (NPTS_F / 4) + TB - 1) / TB, TB, 0, stream>>>(x, assign, ws, n);
  k_fin<<<(MMAX * NPTS_F + TB - 1) / TB, TB, 0, stream>>>(ws, out);
}